// BandEmbedding_6854767804449
// MI455X (gfx1250) — compile-verified
//
#include <hip/hip_runtime.h>
#include <math.h>

#define LN_EPS 1e-5f

typedef __attribute__((ext_vector_type(2))) float v2f;
typedef __attribute__((ext_vector_type(8))) float v8f;

struct AllParams {
    const float* ln_g[3][3];  // [head][layer]
    const float* ln_b[3][3];
    const float* w[3][3];     // [K x 128] row-major
    const float* b[3][3];     // [128]
};

__device__ __forceinline__ float wave_reduce_sum(float v) {
    // wave32 butterfly reduction
    #pragma unroll
    for (int m = 16; m > 0; m >>= 1) v += __shfl_xor(v, m, 32);
    return v;
}

// ---------------------------------------------------------------------------
// One MLP layer: LayerNorm(biased) -> ReLU -> GEMM via WMMA f32 16x16x4.
// K is compile-time so all K-loop guards and trip counts constant-fold:
// the hot loop is pure {ds_load_b64 A-frag, v_wmma} with B resident in VGPRs.
// ---------------------------------------------------------------------------
template <int K>
__device__ __forceinline__ void mlp_layer(float* bufX, float* bufN,
                                          const float* __restrict__ g,
                                          const float* __restrict__ bb,
                                          const float* __restrict__ W,
                                          const float* __restrict__ Bv,
                                          int tid, int lane, int wave,
                                          int lo, int hi, int kk) {
    constexpr int Kpad = (K + 3) & ~3;  // 80 or 128
    constexpr int NK   = Kpad / 4;      // WMMA K-steps

    // zero bufN (zero padding: rows 104..111 and cols K..127)
    for (int i = tid; i < 112 * 128; i += 256) bufN[i] = 0.f;
    __syncthreads();

    // --- LayerNorm + ReLU: one row per wave-iteration ---
    for (int row = wave; row < 104; row += 8) {
        const float* xr = bufX + row * 128;
        const int c0 = lane, c1 = lane + 32, c2 = lane + 64, c3 = lane + 96;
        float v0 = (c0 < K) ? xr[c0] : 0.f;
        float v1 = (c1 < K) ? xr[c1] : 0.f;
        float v2 = (c2 < K) ? xr[c2] : 0.f;
        float v3 = (c3 < K) ? xr[c3] : 0.f;
        float s  = wave_reduce_sum(v0 + v1 + v2 + v3);
        float s2 = wave_reduce_sum(v0 * v0 + v1 * v1 + v2 * v2 + v3 * v3);
        float mean = s / (float)K;
        float var  = s2 / (float)K - mean * mean;
        float inv  = rsqrtf(var + LN_EPS);
        float* nr = bufN + row * 128;
        if (c0 < K) nr[c0] = fmaxf(0.f, (v0 - mean) * inv * g[c0] + bb[c0]);
        if (c1 < K) nr[c1] = fmaxf(0.f, (v1 - mean) * inv * g[c1] + bb[c1]);
        if (c2 < K) nr[c2] = fmaxf(0.f, (v2 - mean) * inv * g[c2] + bb[c2]);
        if (c3 < K) nr[c3] = fmaxf(0.f, (v3 - mean) * inv * g[c3] + bb[c3]);
    }
    __syncthreads();

    // --- preload this wave's B column block (cols n0..n0+15, all K) into VGPRs ---
    const int n0 = wave * 16;
    v2f bf[NK];
    #pragma unroll
    for (int t = 0; t < NK; ++t) {
        const int kr = t * 4 + kk;
        if (t * 4 + 3 < K) {  // constant-folded per unrolled t
            bf[t].x = W[kr * 128 + n0 + lo];
            bf[t].y = W[(kr + 1) * 128 + n0 + lo];
        } else {
            bf[t].x = (kr < K)     ? W[kr * 128 + n0 + lo]       : 0.f;
            bf[t].y = (kr + 1 < K) ? W[(kr + 1) * 128 + n0 + lo] : 0.f;
        }
    }
    const float bias = Bv[n0 + lo];

    // --- GEMM: 7 row tiles, B reused from registers ---
    for (int mt = 0; mt < 7; ++mt) {
        const int m0 = mt * 16;
        const float* ar = bufN + (m0 + lo) * 128 + kk;  // 8B-aligned (kk even)
        v8f acc = {};
        #pragma unroll
        for (int t = 0; t < NK; ++t) {
            v2f a;
            a.x = ar[t * 4];
            a.y = ar[t * 4 + 1];
            acc = __builtin_amdgcn_wmma_f32_16x16x4_f32(
                false, a, false, bf[t], (short)0, acc, false, false);
        }
        #pragma unroll
        for (int v = 0; v < 8; ++v)
            bufX[(m0 + v + 8 * hi) * 128 + n0 + lo] = acc[v] + bias;
    }
    __syncthreads();
}

// ---------------------------------------------------------------------------
// Stage 1: head MLPs, 1 block per head, 256 threads (8 waves).
// Dynamic LDS: bufX[112*128] + bufN[112*128] ping-pong, 112 KB (<320 KB/WGP).
// ---------------------------------------------------------------------------
__global__ __launch_bounds__(256)
void head_mlp_kernel(const float* __restrict__ desc,  // [104][79]
                     AllParams P,
                     float* __restrict__ headOut) {   // [3][104][128]
    extern __shared__ float smem[];
    float* bufX = smem;               // current activations [112][128]
    float* bufN = smem + 112 * 128;   // LN+ReLU output (GEMM A) [112][128]

    const int head = blockIdx.x;
    const int tid  = threadIdx.x;
    const int lane = tid & 31;
    const int wave = tid >> 5;
    const int lo   = lane & 15;
    const int hi   = lane >> 4;
    const int kk   = 2 * hi;

    // bufX <- descriptor, zero-padded to [112][128]
    for (int i = tid; i < 112 * 128; i += 256) bufX[i] = 0.f;
    __syncthreads();
    for (int i = tid; i < 104 * 79; i += 256) {
        int r = i / 79, c = i - r * 79;
        bufX[r * 128 + c] = desc[i];
    }
    __syncthreads();

    mlp_layer<79>(bufX, bufN, P.ln_g[head][0], P.ln_b[head][0],
                  P.w[head][0], P.b[head][0], tid, lane, wave, lo, hi, kk);
    mlp_layer<128>(bufX, bufN, P.ln_g[head][1], P.ln_b[head][1],
                   P.w[head][1], P.b[head][1], tid, lane, wave, lo, hi, kk);
    mlp_layer<128>(bufX, bufN, P.ln_g[head][2], P.ln_b[head][2],
                   P.w[head][2], P.b[head][2], tid, lane, wave, lo, hi, kk);

    for (int i = tid; i < 104 * 128; i += 256)
        headOut[head * 104 * 128 + i] = bufX[i];
}

// ---------------------------------------------------------------------------
// Stage 2: raw embedding (runtime-dominant: ~873M native v_sin_f32, TRANS-bound).
// Coefficients staged transposed in LDS ([e][c], 64B row stride) so the 13
// per-channel f/p/g values merge into ds_load_b128 broadcasts.
// Per thread: 13 band values in registers, 128 coalesced channel stores.
// ---------------------------------------------------------------------------
__global__ __launch_bounds__(256)
void band_sin_kernel(const float* __restrict__ bands,    // [8][13][256][256]
                     const float* __restrict__ headOut,  // [3][104][128]
                     float* __restrict__ out) {          // [8][128][256][256]
    __shared__ float sF[128 * 16];   // [e][c], c-padded to 16 (64B rows)
    __shared__ float sP[128 * 16];
    __shared__ float sG[128 * 16];

    const int b   = blockIdx.x >> 8;                       // 2048 blocks = 8*256
    const int pix = ((blockIdx.x & 255) << 8) + threadIdx.x;

    const float* fb = headOut + 0 * 104 * 128 + b * 13 * 128;
    const float* pb = headOut + 1 * 104 * 128 + b * 13 * 128;
    const float* gb = headOut + 2 * 104 * 128 + b * 13 * 128;
    for (int i = threadIdx.x; i < 13 * 128; i += 256) {
        const int c = i >> 7;          // i = c*128 + e
        const int e = i & 127;
        sF[e * 16 + c] = fb[i];
        sP[e * 16 + c] = pb[i];
        sG[e * 16 + c] = gb[i];
    }
    __syncthreads();

    float band[13];
    #pragma unroll
    for (int c = 0; c < 13; ++c)
        band[c] = bands[((size_t)(b * 13 + c) << 16) + pix];

    float* outp = out + ((size_t)b << 23) + pix;  // b * 128 * 65536
    for (int e = 0; e < 128; ++e) {
        const float* fe = sF + e * 16;
        const float* pe = sP + e * 16;
        const float* ge = sG + e * 16;
        float acc = 0.f;
        #pragma unroll
        for (int c = 0; c < 13; ++c)
            acc += ge[c] * __sinf(fe[c] * (band[c] + pe[c]));
        outp[(size_t)e << 16] = acc;
    }
}

// ---------------------------------------------------------------------------
// Stage 3: deterministic per-channel sum / sumsq over (b,h,w). 1 block per e.
// ---------------------------------------------------------------------------
__global__ __launch_bounds__(256)
void stats_kernel(const float* __restrict__ out, float* __restrict__ stats) {
    const int e = blockIdx.x;
    float s = 0.f, s2 = 0.f;
    for (int b = 0; b < 8; ++b) {
        const float4* p = (const float4*)(out + (((size_t)(b * 128 + e)) << 16));
        for (int i = threadIdx.x; i < 16384; i += 256) {
            float4 v = p[i];
            s  += v.x + v.y + v.z + v.w;
            s2 += v.x * v.x + v.y * v.y + v.z * v.z + v.w * v.w;
        }
    }
    __shared__ float rs[256];
    __shared__ float rq[256];
    rs[threadIdx.x] = s;
    rq[threadIdx.x] = s2;
    __syncthreads();
    for (int off = 128; off > 0; off >>= 1) {
        if (threadIdx.x < off) {
            rs[threadIdx.x] += rs[threadIdx.x + off];
            rq[threadIdx.x] += rq[threadIdx.x + off];
        }
        __syncthreads();
    }
    if (threadIdx.x == 0) {
        stats[e]       = rs[0];
        stats[128 + e] = rq[0];
    }
}

// ---------------------------------------------------------------------------
// Stage 4: BatchNorm (training-mode batch stats), in place, float4 traffic.
// ---------------------------------------------------------------------------
__global__ __launch_bounds__(256)
void bn_kernel(float* __restrict__ out, const float* __restrict__ stats,
               const float* __restrict__ gamma, const float* __restrict__ beta) {
    const size_t i4 = (size_t)blockIdx.x * 256 + threadIdx.x;  // 16777216 float4
    const int e = (int)((i4 >> 14) & 127);                     // 16384 float4 / plane
    const float N = 524288.f;                                  // 8*256*256
    float mean = stats[e] / N;
    float var  = stats[128 + e] / N - mean * mean;
    float inv  = rsqrtf(var + LN_EPS);
    float sc = gamma[e] * inv;
    float sh = beta[e] - mean * sc;
    float4* p = (float4*)out;
    float4 v = p[i4];
    v.x = v.x * sc + sh;
    v.y = v.y * sc + sh;
    v.z = v.z * sc + sh;
    v.w = v.w * sc + sh;
    p[i4] = v;
}

extern "C" void kernel_launch(void* const* d_in, const int* in_sizes, int n_in,
                              void* d_out, int out_size, void* d_ws, size_t ws_size,
                              hipStream_t stream) {
    const float* bands = (const float*)d_in[0];
    const float* desc  = (const float*)d_in[1];

    // dict insertion order per head: ln0_g, ln0_b, w0, b0, ln1_g, ln1_b, w1, b1, ...
    AllParams P;
    for (int h = 0; h < 3; ++h) {
        int base = 2 + h * 12;
        for (int l = 0; l < 3; ++l) {
            P.ln_g[h][l] = (const float*)d_in[base + l * 4 + 0];
            P.ln_b[h][l] = (const float*)d_in[base + l * 4 + 1];
            P.w[h][l]    = (const float*)d_in[base + l * 4 + 2];
            P.b[h][l]    = (const float*)d_in[base + l * 4 + 3];
        }
    }
    const float* gamma = (const float*)d_in[38];
    const float* beta  = (const float*)d_in[39];

    float* out     = (float*)d_out;
    float* ws      = (float*)d_ws;
    float* headOut = ws;                 // 3*104*128 floats
    float* stats   = ws + 3 * 104 * 128; // 256 floats

    head_mlp_kernel<<<3, 256, 2 * 112 * 128 * sizeof(float), stream>>>(desc, P, headOut);
    band_sin_kernel<<<2048, 256, 0, stream>>>(bands, headOut, out);
    stats_kernel<<<128, 256, 0, stream>>>(out, stats);
    bn_kernel<<<65536, 256, 0, stream>>>(out, stats, gamma, beta);
}